// SpectralConv2d_592705487096
// MI455X (gfx1250) — compile-verified
//
#include <hip/hip_runtime.h>

typedef __attribute__((ext_vector_type(2))) float v2f;
typedef __attribute__((ext_vector_type(8))) float v8f;

#define B_   16
#define CIN  64
#define COUT 64
#define H_   256
#define W_   256
#define MH   8
#define MW   12
#define KT   24        // 12 Re cols + 12 Im cols for stage 5

// ---- workspace layout (in floats) ----
#define OFF_FH 0
#define SZ_FH  (16*256)          // forward H basis: rows 0-7 Re=cos/16, rows 8-15 Im=-sin/16
#define OFF_FW (OFF_FH + SZ_FH)
#define SZ_FW  (MW*256*2)        // forward W twiddles, interleaved (re,im), /16
#define OFF_GH (OFF_FW + SZ_FW)
#define SZ_GH  (MH*256*2)        // inverse H twiddles e^{+i}, interleaved, /16
#define OFF_BW (OFF_GH + SZ_GH)
#define SZ_BW  (KT*256)          // C2R basis: rows 0-11 cos-part, rows 12-23 sin-part
#define OFF_A  (OFF_BW + SZ_BW)
#define SZ_A   (B_*CIN*16*256)   // stage-1 output (Re rows 0-7, Im rows 8-15) per (b,i)
#define OFF_X  (OFF_A + SZ_A)
#define SZ_X   (B_*CIN*MH*MW*2)
#define OFF_M  (OFF_X + SZ_X)
#define SZ_M   (B_*COUT*MH*MW*2)
#define OFF_T  (OFF_M + SZ_M)
#define SZ_T   (B_*COUT*H_*KT)   // T tilde: [h][0..11]=Re, [h][12..23]=Im

#ifndef M_PI
#define M_PI 3.14159265358979323846
#endif

// ---------------- K0: build DFT basis tables ----------------
__global__ void spc_tables(float* ws) {
    int t = blockIdx.x * blockDim.x + threadIdx.x;
    const double inv16 = 1.0 / 16.0;            // 1/sqrt(256) per axis (ortho)
    if (t < SZ_FH) {
        int m = t / 256, h = t % 256;
        int ky = (m < 8) ? m : (m - 8);
        int ph = (ky * h) & 255;
        double ang = 2.0 * M_PI * (double)ph / 256.0;
        ws[OFF_FH + t] = (float)((m < 8 ? cos(ang) : -sin(ang)) * inv16);
        return;
    }
    t -= SZ_FH;
    if (t < SZ_FW) {
        int kx = t / 512, r = t % 512, w = r >> 1, c = r & 1;
        int ph = (kx * w) & 255;
        double ang = 2.0 * M_PI * (double)ph / 256.0;
        ws[OFF_FW + kx*512 + r] = (float)((c == 0 ? cos(ang) : -sin(ang)) * inv16);
        return;
    }
    t -= SZ_FW;
    if (t < SZ_GH) {
        int ky = t / 512, r = t % 512, h = r >> 1, c = r & 1;
        int ph = (ky * h) & 255;
        double ang = 2.0 * M_PI * (double)ph / 256.0;
        ws[OFF_GH + ky*512 + r] = (float)((c == 0 ? cos(ang) : sin(ang)) * inv16);
        return;
    }
    t -= SZ_GH;
    if (t < SZ_BW) {
        int row = t / 256, w = t % 256;
        double v;
        if (row < 12) {                 // cos-part coefficients
            int kx = row;
            if (kx == 0) v = inv16;
            else {
                int ph = (kx * w) & 255;
                v = 2.0 * cos(2.0 * M_PI * (double)ph / 256.0) * inv16;
            }
        } else {                        // sin-part coefficients (Im of kx=0 bin dropped)
            int kx = row - 12;
            if (kx == 0) v = 0.0;
            else {
                int ph = (kx * w) & 255;
                v = -2.0 * sin(2.0 * M_PI * (double)ph / 256.0) * inv16;
            }
        }
        ws[OFF_BW + t] = (float)v;
        return;
    }
}

// ---------------- K1: stage 1 — A = Fh(16x256) * x(256x256) per (b,i), WMMA f32 ----------------
__global__ void spc_stage1(const float* __restrict__ x, float* __restrict__ ws) {
    int wave = blockIdx.x * (blockDim.x >> 5) + (threadIdx.x >> 5);
    int lane = threadIdx.x & 31;
    int lrow = lane & 15;
    int lhalf = lane >> 4;
    int bi = wave >> 4;          // which (b, c_in)
    int wt = wave & 15;          // 16-wide tile along W

    const float* xb = x + (size_t)bi * (H_ * W_);
    const float* Fh = ws + OFF_FH;
    int n = wt * 16 + lrow;

    v8f acc = {};
#pragma unroll 4
    for (int kk = 0; kk < 64; ++kk) {
        int k = kk * 4 + lhalf * 2;
        v2f a = *(const v2f*)(Fh + lrow * 256 + k);         // A[m=lrow][k..k+1]
        v2f b;
        b.x = xb[k * W_ + n];                               // B[k][n]
        b.y = xb[(k + 1) * W_ + n];                         // B[k+1][n]
        acc = __builtin_amdgcn_wmma_f32_16x16x4_f32(
            false, a, false, b, (short)0, acc, false, false);
    }
    float* Ab = ws + OFF_A + (size_t)bi * (16 * 256);
#pragma unroll
    for (int r = 0; r < 8; ++r)
        Ab[(r + lhalf * 8) * 256 + wt * 16 + lrow] = acc[r];
}

// ---------------- K2: stage 2 — project A onto 12 W-modes ----------------
__global__ void spc_stage2(float* __restrict__ ws) {
    int t = blockIdx.x * blockDim.x + threadIdx.x;   // B_*CIN*96 threads
    int bi = t / (MH * MW);
    int mode = t % (MH * MW);
    int ky = mode / MW, kx = mode % MW;
    const float* Are = ws + OFF_A + (size_t)bi * (16 * 256) + ky * 256;
    const float* Aim = Are + 8 * 256;
    const float* Fw  = ws + OFF_FW + kx * 512;
    float xre = 0.f, xim = 0.f;
    for (int w = 0; w < 256; ++w) {
        float fre = Fw[w * 2], fim = Fw[w * 2 + 1];
        float ar = Are[w], ai = Aim[w];
        xre += ar * fre - ai * fim;
        xim += ar * fim + ai * fre;
    }
    float* X = ws + OFF_X + (size_t)bi * (MH * MW * 2) + mode * 2;
    X[0] = xre; X[1] = xim;
}

// ---------------- K3: stage 3 — complex channel mixing with weight ----------------
__global__ void spc_stage3(const float* __restrict__ weight, float* __restrict__ ws) {
    int t = blockIdx.x * blockDim.x + threadIdx.x;   // B_*COUT*96 threads
    int mode = t % (MH * MW);
    int bo = t / (MH * MW);
    int o = bo % COUT;
    int b = bo / COUT;
    float mre = 0.f, mim = 0.f;
    for (int i = 0; i < CIN; ++i) {
        const float* X = ws + OFF_X + (size_t)(b * CIN + i) * (MH * MW * 2) + mode * 2;
        const float* Wp = weight + ((size_t)(i * COUT + o) * (MH * MW) + mode) * 2;
        float xr = X[0], xi = X[1];
        float wr = Wp[0], wi = Wp[1];
        mre += xr * wr - xi * wi;
        mim += xr * wi + xi * wr;
    }
    float* M = ws + OFF_M + (size_t)bo * (MH * MW * 2) + mode * 2;
    M[0] = mre; M[1] = mim;
}

// ---------------- K4: stage 4 — inverse DFT over H: T[h][kx] ----------------
__global__ void spc_stage4(float* __restrict__ ws) {
    __shared__ float sM[MH * MW * 2];
    int bo = blockIdx.x;                 // (b, c_out)
    int h = threadIdx.x;                 // 256 threads
    if (h < MH * MW * 2 - 64) { }        // (no-op; keep full EXEC)
    if (threadIdx.x < MH * MW * 2)
        sM[threadIdx.x] = ws[OFF_M + (size_t)bo * (MH * MW * 2) + threadIdx.x];
    __syncthreads();

    const float* Gh = ws + OFF_GH;
    float* T = ws + OFF_T + (size_t)bo * (H_ * KT) + h * KT;
#pragma unroll
    for (int kx = 0; kx < MW; ++kx) {
        float tre = 0.f, tim = 0.f;
#pragma unroll
        for (int ky = 0; ky < MH; ++ky) {
            float mre = sM[(ky * MW + kx) * 2];
            float mim = sM[(ky * MW + kx) * 2 + 1];
            float gre = Gh[(ky * 256 + h) * 2];
            float gim = Gh[(ky * 256 + h) * 2 + 1];
            tre += mre * gre - mim * gim;
            tim += mre * gim + mim * gre;
        }
        T[kx] = tre;
        T[12 + kx] = tim;
    }
}

// ---------------- K5: stage 5 — out = Ttilde(256x24) * Bw(24x256) per (b,o), WMMA f32 ----------------
__global__ void spc_stage5(const float* __restrict__ ws, float* __restrict__ out) {
    int wave = blockIdx.x * (blockDim.x >> 5) + (threadIdx.x >> 5);
    int lane = threadIdx.x & 31;
    int lrow = lane & 15;
    int lhalf = lane >> 4;
    int bo = wave >> 8;                  // (b, c_out)
    int tile = wave & 255;
    int h0 = (tile >> 4) * 16;
    int w0 = (tile & 15) * 16;

    const float* T = ws + OFF_T + (size_t)bo * (H_ * KT);
    const float* Bw = ws + OFF_BW;
    int n = w0 + lrow;

    v8f acc = {};
#pragma unroll
    for (int kk = 0; kk < 6; ++kk) {
        int k = kk * 4 + lhalf * 2;
        v2f a = *(const v2f*)(T + (h0 + lrow) * KT + k);    // A[m][k..k+1]
        v2f b;
        b.x = Bw[k * 256 + n];
        b.y = Bw[(k + 1) * 256 + n];
        acc = __builtin_amdgcn_wmma_f32_16x16x4_f32(
            false, a, false, b, (short)0, acc, false, false);
    }
    float* ob = out + (size_t)bo * (H_ * W_);
#pragma unroll
    for (int r = 0; r < 8; ++r)
        ob[(h0 + r + lhalf * 8) * W_ + w0 + lrow] = acc[r];
}

extern "C" void kernel_launch(void* const* d_in, const int* in_sizes, int n_in,
                              void* d_out, int out_size, void* d_ws, size_t ws_size,
                              hipStream_t stream) {
    const float* x = (const float*)d_in[0];        // (16,64,256,256) f32
    const float* weight = (const float*)d_in[1];   // (64,64,8,12,2) f32
    float* out = (float*)d_out;                    // (16,64,256,256) f32
    float* ws = (float*)d_ws;

    // K0: tables (20480 entries)
    {
        int total = SZ_FH + SZ_FW + SZ_GH + SZ_BW;
        spc_tables<<<(total + 255) / 256, 256, 0, stream>>>(ws);
    }
    // K1: forward H-DFT via WMMA f32: 16384 waves -> 2048 blocks of 8 waves
    spc_stage1<<<(B_ * CIN * 16) / 8, 256, 0, stream>>>(x, ws);
    // K2: W-mode projection (B*CIN*96 threads)
    spc_stage2<<<(B_ * CIN * MH * MW) / 256, 256, 0, stream>>>(ws);
    // K3: channel mixing (B*COUT*96 threads)
    spc_stage3<<<(B_ * COUT * MH * MW) / 256, 256, 0, stream>>>(weight, ws);
    // K4: inverse H-DFT (one block per (b,o), 256 threads = h)
    spc_stage4<<<B_ * COUT, 256, 0, stream>>>(ws);
    // K5: C2R reconstruction via WMMA f32: 262144 waves -> 32768 blocks
    spc_stage5<<<(B_ * COUT * 16 * 16) / 8, 256, 0, stream>>>(ws, out);
}